// RealtimeNgramProcessor_25890062860790
// MI455X (gfx1250) — compile-verified
//
#include <hip/hip_runtime.h>

// RealtimeNgramProcessor: encode 2-/3-gram windows (17 bits/token) and look
// them up in sorted int64 key tables (searchsorted + verify), output [2,B,S].
// Memory-bound integer search workload: no matrix structure -> no WMMA.
// Strategy:
//   * LDS first-level index (1 key per 64-key segment, 12.5KB/table) filled
//     with global_load_async_to_lds_b64 (ASYNCcnt path).
//   * Branchless uniform-trip binary search: top level in LDS (scalar loop
//     control, ds_load + v_cndmask per step), leaf level in a 512B segment
//     that is L2-resident (3.2MB of tables vs 192MB L2).
//   * ~97% of lookups miss (100K keys of ~4M unique n-grams), so vals[] is
//     only touched under the hit predicate; one prefetch warms the leaf line.
// Compulsory HBM traffic ~66MB -> ~3us floor at 23.3 TB/s.

typedef long long i64;
typedef i64   v2i64 __attribute__((ext_vector_type(2)));
typedef float v4f   __attribute__((ext_vector_type(4)));
typedef int   v2i32 __attribute__((ext_vector_type(2)));

// Address-space-qualified pointee types for the async global->LDS builtin
// (param type per hipcc diagnostic: pointer to 2xi32 vector).
typedef __attribute__((address_space(1))) v2i32 gas_v2i32;
typedef __attribute__((address_space(3))) v2i32 las_v2i32;

#define BB        256      // batch
#define SS        16384    // sequence length
#define TPB       256      // 8 wave32 per block
#define PPT       16       // positions per thread
#define SEG_LOG   6
#define SEG       64       // keys per leaf segment (512 B)
#define MAXSEG    2048     // supports up to 131072 keys per table
#define BASE_BITS 17

#if __has_builtin(__builtin_amdgcn_global_load_async_to_lds_b64) && \
    __has_builtin(__builtin_amdgcn_s_wait_asynccnt)
#define HAVE_ASYNC_LDS 1
#else
#define HAVE_ASYNC_LDS 0
#endif

__device__ __forceinline__ float ngram_lookup(i64 v,
                                              const i64* __restrict__ keys,
                                              const i64* __restrict__ vals,
                                              int K,
                                              const i64* __restrict__ sIdx,
                                              int nseg)
{
    // ---- Top level (LDS): branchless upper_bound over sampled keys. ----
    // n is wave-uniform (same nseg for all lanes) -> scalar loop control;
    // per-lane work is ds_load_b64 + compare + cndmask per step.
    int base = 0, n = nseg;
    while (n > 1) {
        int half = n >> 1;
        base = (sIdx[base + half] <= v) ? base + half : base;
        n -= half;
    }
    int seg = base + ((sIdx[base] <= v) ? 1 : 0) - 1;
    seg = max(seg, 0);                 // v < keys[0]: verify below rejects
    int st = seg << SEG_LOG;
    int en = min(st + SEG, K);

    // Warm line 0 of the leaf segment while probe 0 (mid-segment) is issued.
    __builtin_prefetch(&keys[st], 0, 3);

    // ---- Leaf level: branchless lower_bound within the 512B segment. ----
    int b2 = st, n2 = en - st;         // n2 == 64 except in the last segment
    while (n2 > 1) {
        int half = n2 >> 1;
        b2 = (keys[b2 + half] < v) ? b2 + half : b2;
        n2 -= half;
    }
    int l   = b2 + ((keys[b2] < v) ? 1 : 0);
    int idx = min(l, en - 1);          // safe reload index (same cache line)
    if (l < en && keys[idx] == v)
        return (float)vals[idx];       // vals touched only on (rare) hit
    return 0.0f;
}

__global__ __launch_bounds__(TPB)
void ngram_kernel(const i64* __restrict__ x,
                  const i64* __restrict__ keys2, const i64* __restrict__ vals2, int K2,
                  const i64* __restrict__ keys3, const i64* __restrict__ vals3, int K3,
                  float* __restrict__ out)
{
    __shared__ i64 sIdx2[MAXSEG];
    __shared__ i64 sIdx3[MAXSEG];
    const int nseg2 = (K2 + SEG - 1) >> SEG_LOG;
    const int nseg3 = (K3 + SEG - 1) >> SEG_LOG;

#if HAVE_ASYNC_LDS
    // Async global->LDS fill of the first-level index (ASYNCcnt path,
    // global_load_async_to_lds_b64).
    for (int j = threadIdx.x; j < nseg2; j += TPB)
        __builtin_amdgcn_global_load_async_to_lds_b64(
            (gas_v2i32*)(keys2 + ((size_t)j << SEG_LOG)),
            (las_v2i32*)(sIdx2 + j), 0, 0);
    for (int j = threadIdx.x; j < nseg3; j += TPB)
        __builtin_amdgcn_global_load_async_to_lds_b64(
            (gas_v2i32*)(keys3 + ((size_t)j << SEG_LOG)),
            (las_v2i32*)(sIdx3 + j), 0, 0);
    __builtin_amdgcn_s_wait_asynccnt(0);
    __syncthreads();
#else
    for (int j = threadIdx.x; j < nseg2; j += TPB)
        sIdx2[j] = keys2[(size_t)j << SEG_LOG];
    for (int j = threadIdx.x; j < nseg3; j += TPB)
        sIdx3[j] = keys3[(size_t)j << SEG_LOG];
    __syncthreads();
#endif

    const int tid           = blockIdx.x * TPB + threadIdx.x;
    const int threadsPerRow = SS / PPT;            // 1024
    const int b  = tid / threadsPerRow;
    const int s0 = (tid % threadsPerRow) * PPT;

    // Load 16 tokens (b128-vectorized) + 2 history tokens (zero-padded).
    const i64* row = x + (size_t)b * SS + s0;
    i64 tok[PPT + 2];
    tok[0] = (s0 >= 2) ? row[-2] : 0;
    tok[1] = (s0 >= 1) ? row[-1] : 0;
#pragma unroll
    for (int i = 0; i < PPT / 2; ++i) {
        v2i64 t = ((const v2i64*)row)[i];          // global_load_b128
        tok[2 + 2 * i]     = t.x;
        tok[2 + 2 * i + 1] = t.y;
    }

    float r2[PPT], r3[PPT];
#pragma unroll
    for (int i = 0; i < PPT; ++i) {
        i64 e2 = (tok[i + 1] << BASE_BITS) | tok[i + 2];        // (t[j-1],t[j])
        i64 e3 = (tok[i] << (2 * BASE_BITS)) | e2;              // (t[j-2],t[j-1],t[j])
        r2[i] = ngram_lookup(e2, keys2, vals2, K2, sIdx2, nseg2);
        r3[i] = ngram_lookup(e3, keys3, vals3, K3, sIdx3, nseg3);
    }

    // Plane 0 = 2-grams, plane 1 = 3-grams; float4 stores (16B aligned).
    float* o2 = out + (size_t)b * SS + s0;
    float* o3 = out + (size_t)BB * SS + (size_t)b * SS + s0;
#pragma unroll
    for (int i = 0; i < PPT / 4; ++i) {
        v4f a = { r2[4 * i], r2[4 * i + 1], r2[4 * i + 2], r2[4 * i + 3] };
        v4f c = { r3[4 * i], r3[4 * i + 1], r3[4 * i + 2], r3[4 * i + 3] };
        ((v4f*)o2)[i] = a;
        ((v4f*)o3)[i] = c;
    }
}

extern "C" void kernel_launch(void* const* d_in, const int* in_sizes, int n_in,
                              void* d_out, int out_size, void* d_ws, size_t ws_size,
                              hipStream_t stream) {
    const i64* x     = (const i64*)d_in[0];
    const i64* keys2 = (const i64*)d_in[1];
    const i64* vals2 = (const i64*)d_in[2];
    const i64* keys3 = (const i64*)d_in[3];
    const i64* vals3 = (const i64*)d_in[4];
    const int  K2 = in_sizes[1];
    const int  K3 = in_sizes[3];

    dim3 grid((BB * SS) / (TPB * PPT));            // 1024 blocks
    dim3 block(TPB);
    ngram_kernel<<<grid, block, 0, stream>>>(x, keys2, vals2, K2,
                                             keys3, vals3, K3,
                                             (float*)d_out);
}